// RGCN_node_42116449304687
// MI455X (gfx1250) — compile-verified
//
#include <hip/hip_runtime.h>
#include <hip/hip_bf16.h>

typedef __attribute__((ext_vector_type(2))) float v2f;
typedef __attribute__((ext_vector_type(8))) float v8f;

#define HDIM 128
#define RREL 8

// ---------------------------------------------------------------------------
// Degree: deg[n*R + r] = count of incoming edges of relation r at node n
// ---------------------------------------------------------------------------
__global__ void deg_kernel(const int* __restrict__ dst, const int* __restrict__ et,
                           float* __restrict__ deg, int E) {
    int e = blockIdx.x * blockDim.x + threadIdx.x;
    if (e >= E) return;
    atomicAdd(&deg[dst[e] * RREL + et[e]], 1.0f);
}

// ---------------------------------------------------------------------------
// WMMA GEMM: OUT[16-row tile, 0..127] = X[tile rows, 0..K-1] @ W[K,128] (+bias)
// One wave per 16-row tile; f32 WMMA 16x16x4, K fully unrolled, A in registers.
// ---------------------------------------------------------------------------
template <int K>
__global__ void gemm16(const float* __restrict__ X, const float* __restrict__ W,
                       const float* __restrict__ bias, float* __restrict__ OUT,
                       int ntiles) {
    constexpr int K4 = K / 4;
    const int lane = threadIdx.x & 31;
    const int wv   = threadIdx.x >> 5;
    const int tile = blockIdx.x * (blockDim.x >> 5) + wv;
    if (tile >= ntiles) return;                 // wave-uniform: EXEC all-1s below

    const int l16 = lane & 15;                  // A: row index | B/C/D: col index
    const int hi  = lane >> 4;                  // half-wave selector

    // Preload A fragments: lane holds row (tile*16+l16), K pair 2*hi .. 2*hi+1 per step
    const float* xr = X + (size_t)(tile * 16 + l16) * K;
    v2f a[K4];
#pragma unroll
    for (int kk = 0; kk < K4; ++kk) {
        const int k = kk * 4 + hi * 2;
        a[kk].x = xr[k];
        a[kk].y = xr[k + 1];
    }

    for (int ht = 0; ht < HDIM / 16; ++ht) {
        const int col = ht * 16 + l16;
        const float binit = bias ? bias[col] : 0.0f;
        v8f acc = {binit, binit, binit, binit, binit, binit, binit, binit};
#pragma unroll
        for (int kk = 0; kk < K4; ++kk) {
            const int k = kk * 4 + hi * 2;
            v2f b;
            b.x = W[(size_t)k * HDIM + col];
            b.y = W[(size_t)(k + 1) * HDIM + col];
            acc = __builtin_amdgcn_wmma_f32_16x16x4_f32(
                false, a[kk], false, b, (short)0, acc, false, false);
        }
        // Store: lane covers column `col`, rows tile*16 + 8*hi + v
        float* orow = OUT + (size_t)(tile * 16 + hi * 8) * HDIM + col;
#pragma unroll
        for (int v = 0; v < 8; ++v) orow[(size_t)v * HDIM] = acc[v];
    }
}

// ---------------------------------------------------------------------------
// Edge scatter for one relation: pre[dst] += xw_r[src] * (1/max(deg,1))
// One wave per edge; lane moves 4 channels (float4) of the 128.
// ---------------------------------------------------------------------------
__global__ void scatter_rel(const float* __restrict__ xw, float* __restrict__ pre,
                            const int* __restrict__ src, const int* __restrict__ dst,
                            const int* __restrict__ et, const float* __restrict__ deg,
                            int E, int r) {
    const int e    = blockIdx.x * (blockDim.x >> 5) + (threadIdx.x >> 5);
    const int lane = threadIdx.x & 31;
    if (e >= E) return;
    if (et[e] != r) return;                      // wave-uniform branch
    const int s = src[e];
    const int d = dst[e];
    const float norm = 1.0f / fmaxf(deg[d * RREL + r], 1.0f);
    const float4 v = ((const float4*)(xw + (size_t)s * HDIM))[lane];
    float* p = pre + (size_t)d * HDIM + lane * 4;
    atomicAdd(p + 0, v.x * norm);
    atomicAdd(p + 1, v.y * norm);
    atomicAdd(p + 2, v.z * norm);
    atomicAdd(p + 3, v.w * norm);
}

// ---------------------------------------------------------------------------
// BatchNorm statistics / finalize / apply(+ReLU, in place)
// ---------------------------------------------------------------------------
__global__ void bn_stats(const float* __restrict__ pre, float* __restrict__ gsum,
                         float* __restrict__ gsq, int N) {
    const int ch = threadIdx.x;                  // 128 channels
    float s = 0.0f, sq = 0.0f;
    for (int n = blockIdx.x; n < N; n += gridDim.x) {
        const float v = pre[(size_t)n * HDIM + ch];
        s += v;
        sq += v * v;
    }
    atomicAdd(&gsum[ch], s);
    atomicAdd(&gsq[ch], sq);
}

__global__ void bn_finalize(const float* __restrict__ gsum, const float* __restrict__ gsq,
                            const float* __restrict__ gamma, const float* __restrict__ beta,
                            float* __restrict__ sc, float* __restrict__ sh, int N) {
    const int ch = threadIdx.x;
    const float inv_n = 1.0f / (float)N;
    const float mu  = gsum[ch] * inv_n;
    const float var = gsq[ch] * inv_n - mu * mu;
    const float s   = gamma[ch] * rsqrtf(var + 1e-5f);
    sc[ch] = s;
    sh[ch] = beta[ch] - mu * s;
}

__global__ void bn_apply_relu(float* __restrict__ h, const float* __restrict__ sc,
                              const float* __restrict__ sh, int total) {
    const int i = blockIdx.x * blockDim.x + threadIdx.x;
    if (i >= total) return;
    const int ch = i & (HDIM - 1);
    h[i] = fmaxf(h[i] * sc[ch] + sh[ch], 0.0f);
}

// ---------------------------------------------------------------------------
// Final linear: out[n,c] = h[n,:] @ Wf[:,c] + bf[c]   (C = 8)
// ---------------------------------------------------------------------------
__global__ void final_linear(const float* __restrict__ h, const float* __restrict__ Wf,
                             const float* __restrict__ bf, float* __restrict__ out, int N) {
    const int i = blockIdx.x * blockDim.x + threadIdx.x;   // n*8 + c
    if (i >= N * 8) return;
    const int n = i >> 3;
    const int c = i & 7;
    const float* row = h + (size_t)n * HDIM;
    float acc = bf[c];
#pragma unroll
    for (int k = 0; k < HDIM; ++k) acc = fmaf(row[k], Wf[k * 8 + c], acc);
    out[i] = acc;
}

// ---------------------------------------------------------------------------
// Launch
// ---------------------------------------------------------------------------
extern "C" void kernel_launch(void* const* d_in, const int* in_sizes, int n_in,
                              void* d_out, int out_size, void* d_ws, size_t ws_size,
                              hipStream_t stream) {
    const float* x       = (const float*)d_in[0];
    const int*   src     = (const int*)d_in[1];
    const int*   dst     = (const int*)d_in[2];
    const int*   et      = (const int*)d_in[3];
    const float* W1_rel  = (const float*)d_in[4];   // [8,64,128]
    const float* W1_root = (const float*)d_in[5];   // [64,128]
    const float* b1      = (const float*)d_in[6];
    const float* g1      = (const float*)d_in[7];
    const float* be1     = (const float*)d_in[8];
    const float* W2_rel  = (const float*)d_in[9];   // [8,128,128]
    const float* W2_root = (const float*)d_in[10];  // [128,128]
    const float* b2      = (const float*)d_in[11];
    const float* g2      = (const float*)d_in[12];
    const float* be2     = (const float*)d_in[13];
    const float* Wf      = (const float*)d_in[14];  // [128,8]
    const float* bf      = (const float*)d_in[15];
    float* out = (float*)d_out;

    const int F = 64;
    const int N = in_sizes[0] / F;
    const int E = in_sizes[1];
    const int ntiles = (N + 15) / 16;

    // Workspace layout (floats)
    float* deg  = (float*)d_ws;                  // N*R
    float* xw   = deg  + (size_t)N * RREL;       // N*128 (one relation at a time)
    float* pre1 = xw   + (size_t)N * HDIM;       // N*128
    float* pre2 = pre1 + (size_t)N * HDIM;       // N*128
    float* gsum = pre2 + (size_t)N * HDIM;       // 128
    float* gsq  = gsum + HDIM;                   // 128
    float* sc   = gsq  + HDIM;                   // 128
    float* sh   = sc   + HDIM;                   // 128

    const int T256 = 256;
    const dim3 gemm_blk(128);                                 // 4 waves
    const dim3 gemm_grd((ntiles + 3) / 4);
    const dim3 scat_grd((E + 7) / 8);                         // 8 edges/block

    // degrees (shared by both layers)
    hipMemsetAsync(deg, 0, (size_t)N * RREL * sizeof(float), stream);
    deg_kernel<<<(E + T256 - 1) / T256, T256, 0, stream>>>(dst, et, deg, E);

    // ---- Layer 1 ----
    gemm16<64><<<gemm_grd, gemm_blk, 0, stream>>>(x, W1_root, b1, pre1, ntiles);
    for (int r = 0; r < RREL; ++r) {
        gemm16<64><<<gemm_grd, gemm_blk, 0, stream>>>(x, W1_rel + (size_t)r * F * HDIM,
                                                      nullptr, xw, ntiles);
        scatter_rel<<<scat_grd, T256, 0, stream>>>(xw, pre1, src, dst, et, deg, E, r);
    }
    hipMemsetAsync(gsum, 0, 2 * HDIM * sizeof(float), stream);
    bn_stats<<<512, HDIM, 0, stream>>>(pre1, gsum, gsq, N);
    bn_finalize<<<1, HDIM, 0, stream>>>(gsum, gsq, g1, be1, sc, sh, N);
    bn_apply_relu<<<((size_t)N * HDIM + T256 - 1) / T256, T256, 0, stream>>>(
        pre1, sc, sh, N * HDIM);

    // ---- Layer 2 ----
    gemm16<128><<<gemm_grd, gemm_blk, 0, stream>>>(pre1, W2_root, b2, pre2, ntiles);
    for (int r = 0; r < RREL; ++r) {
        gemm16<128><<<gemm_grd, gemm_blk, 0, stream>>>(pre1, W2_rel + (size_t)r * HDIM * HDIM,
                                                       nullptr, xw, ntiles);
        scatter_rel<<<scat_grd, T256, 0, stream>>>(xw, pre2, src, dst, et, deg, E, r);
    }
    hipMemsetAsync(gsum, 0, 2 * HDIM * sizeof(float), stream);
    bn_stats<<<512, HDIM, 0, stream>>>(pre2, gsum, gsq, N);
    bn_finalize<<<1, HDIM, 0, stream>>>(gsum, gsq, g2, be2, sc, sh, N);
    bn_apply_relu<<<((size_t)N * HDIM + T256 - 1) / T256, T256, 0, stream>>>(
        pre2, sc, sh, N * HDIM);

    // ---- Final linear ----
    final_linear<<<((size_t)N * 8 + T256 - 1) / T256, T256, 0, stream>>>(pre2, Wf, bf, out, N);
}